// DecoderPart2Transformer_20315195310482
// MI455X (gfx1250) — compile-verified
//
#include <hip/hip_runtime.h>
#include <hip/hip_bf16.h>
#include <math.h>

// ---- problem constants (match reference) ----
#define LL   8
#define BB   8
#define NN   4
#define HIN  1024
#define DD   1024
#define NHH  16
#define KVHH 8
#define HDD  64
#define FFF  3072
#define WW   128
#define DOUTT 1024
#define MM   32            // B*N rows of activations
#define EPSF 1e-6f

typedef __attribute__((ext_vector_type(2))) float v2f;
typedef __attribute__((ext_vector_type(8))) float v8f;

// ============================================================
// WMMA fp32 GEMM: Y[32,Nc] += X[32,K] @ W[K,Nc]
// grid = (Nc/64, K/128), block = 128 threads (4 waves).
// Each wave: one 16-col strip, both 16-row M tiles.
// X tile staged global->LDS with GLOBAL_LOAD_ASYNC_TO_LDS_B128
// (ASYNCcnt); weights streamed non-temporally straight into B
// fragments; split-K partials via global_atomic_add_f32.
// ============================================================
__global__ __launch_bounds__(128) void gemm32_wmma_f32(
    const float* __restrict__ X, const float* __restrict__ Wm,
    float* __restrict__ Y, int K, int Nc)
{
    __shared__ float Xs[32][132];             // pad stride 132 -> conflict-free
    const int lane = threadIdx.x & 31;
    const int wv   = threadIdx.x >> 5;
    const int half = lane >> 4;               // 0: lanes 0-15, 1: lanes 16-31
    const int lr   = lane & 15;
    const int n0   = blockIdx.x * 64 + wv * 16;
    const int kb   = blockIdx.y * 128;

    // async copy X[0:32, kb:kb+128] -> LDS: 16 B per lane per op,
    // 1024 x 16B chunks = 128 threads x 8 ops. Fully coalesced, 16B aligned.
    {
        const unsigned ldsBase = (unsigned)(size_t)(&Xs[0][0]);
#pragma unroll
        for (int i = 0; i < 8; ++i) {
            int c   = i * 128 + threadIdx.x;   // vec4 chunk id
            int r   = c >> 5;                  // row 0..31 (32 vec4 per row)
            int col = (c & 31) << 2;           // float col, multiple of 4
            unsigned lds = ldsBase + (unsigned)(r * 132 + col) * 4u;
            const float* gp = X + (size_t)r * K + kb + col;
            asm volatile("global_load_async_to_lds_b128 %0, %1, off"
                         :: "v"(lds), "v"(gp) : "memory");
        }
        asm volatile("s_wait_asynccnt 0x0" ::: "memory");
    }
    __syncthreads();

    v8f acc0 = {};   // rows 0..15
    v8f acc1 = {};   // rows 16..31
    // strength-reduced weight pointer: starts at row (kb + 2*half), col n0+lr
    const float* wq = Wm + (size_t)(kb + 2 * half) * Nc + n0 + lr;

#pragma unroll 8
    for (int k4 = 0; k4 < 128; k4 += 4) {
        // A fragments (16x4 fp32): VGPR0 = K{0,2}, VGPR1 = K{1,3} per half-wave
        v2f a0, a1, b;
        a0.x = Xs[lr][k4 + 2 * half];
        a0.y = Xs[lr][k4 + 2 * half + 1];
        a1.x = Xs[lr + 16][k4 + 2 * half];
        a1.y = Xs[lr + 16][k4 + 2 * half + 1];
        // B fragment (4x16 fp32), streamed non-temporally (single-use weights)
        b.x = __builtin_nontemporal_load(wq);
        b.y = __builtin_nontemporal_load(wq + Nc);
        wq += (size_t)4 * Nc;
        acc0 = __builtin_amdgcn_wmma_f32_16x16x4_f32(false, a0, false, b,
                                                     (short)0, acc0, false, false);
        acc1 = __builtin_amdgcn_wmma_f32_16x16x4_f32(false, a1, false, b,
                                                     (short)0, acc1, false, false);
    }

    const int col = n0 + lr;
#pragma unroll
    for (int r = 0; r < 8; ++r) {
        int row = r + 8 * half;               // C/D layout: VGPRr = M{r, r+8}
        atomicAdd(&Y[(size_t)row * Nc + col], acc0[r]);
        atomicAdd(&Y[(size_t)(row + 16) * Nc + col], acc1[r]);
    }
}

// ============================================================
// fill (zero-init before split-K atomic accumulation)
// ============================================================
__global__ void fillf(float* __restrict__ p, float v, int n)
{
    int i = blockIdx.x * 256 + threadIdx.x;
    if (i < n) p[i] = v;
}

// ============================================================
// RMSNorm over 32 rows of length Dd:  out = in * rsqrt(mean(in^2)+eps) * w
// ============================================================
__global__ __launch_bounds__(256) void rmsnorm32(
    const float* __restrict__ in, const float* __restrict__ w,
    float* __restrict__ out, int Dd)
{
    __shared__ float red[256];
    const int row = blockIdx.x;
    const float* p = in + (size_t)row * Dd;
    float s = 0.f;
    for (int d = threadIdx.x; d < Dd; d += 256) { float v = p[d]; s += v * v; }
    red[threadIdx.x] = s;
    __syncthreads();
    for (int off = 128; off > 0; off >>= 1) {
        if (threadIdx.x < off) red[threadIdx.x] += red[threadIdx.x + off];
        __syncthreads();
    }
    const float scale = rsqrtf(red[0] / (float)Dd + EPSF);
    float* q = out + (size_t)row * Dd;
    for (int d = threadIdx.x; d < Dd; d += 256) q[d] = p[d] * scale * w[d];
}

// ============================================================
// Per-head RMS (qn) + RoPE for Q, in place. grid = 32*16 heads, 64 thr.
// row = (b*4+n)*16 + h ; position = W + n
// ============================================================
__global__ __launch_bounds__(64) void rope_q(
    float* __restrict__ q, const float* __restrict__ qn)
{
    __shared__ float sh[64];
    __shared__ float sv[64];
    const int row = blockIdx.x;
    const int n   = (row >> 4) & 3;
    const int t   = threadIdx.x;
    float* p = q + (size_t)row * HDD;
    float v = p[t];
    sh[t] = v * v;
    __syncthreads();
    for (int off = 32; off > 0; off >>= 1) {
        if (t < off) sh[t] += sh[t + off];
        __syncthreads();
    }
    const float scale = rsqrtf(sh[0] / (float)HDD + EPSF);
    const float xs = v * scale * qn[t];
    sv[t] = xs;
    __syncthreads();
    const int i = t & 31;
    const float ang = (float)(WW + n) *
                      __expf(-((float)(2 * i) / (float)HDD) * __logf(1.0e6f));
    const float xr = (t < 32) ? -sv[t + 32] : sv[t - 32];
    p[t] = xs * __cosf(ang) + xr * __sinf(ang);
}

// ============================================================
// Per-head RMS (kn) + RoPE for K, written straight into the KV-cache tail,
// plus V tail copy. grid = 32*8, 64 thr. row = (b*4+n)*8 + kvh
// ============================================================
__global__ __launch_bounds__(64) void rope_kv(
    const float* __restrict__ kbuf, const float* __restrict__ vbuf,
    const float* __restrict__ kn, float* __restrict__ okl, float* __restrict__ ovl)
{
    __shared__ float sh[64];
    __shared__ float sv[64];
    const int row = blockIdx.x;
    const int b   = row >> 5;
    const int n   = (row >> 3) & 3;
    const int kvh = row & 7;
    const int t   = threadIdx.x;
    const float* p = kbuf + (size_t)row * HDD;
    float v = p[t];
    sh[t] = v * v;
    __syncthreads();
    for (int off = 32; off > 0; off >>= 1) {
        if (t < off) sh[t] += sh[t + off];
        __syncthreads();
    }
    const float scale = rsqrtf(sh[0] / (float)HDD + EPSF);
    const float xs = v * scale * kn[t];
    sv[t] = xs;
    __syncthreads();
    const int i = t & 31;
    const float ang = (float)(WW + n) *
                      __expf(-((float)(2 * i) / (float)HDD) * __logf(1.0e6f));
    const float xr = (t < 32) ? -sv[t + 32] : sv[t - 32];
    const size_t base = (((size_t)b * KVHH + kvh) * WW + (WW - NN + n)) * HDD;
    okl[base + t] = xs * __cosf(ang) + xr * __sinf(ang);
    ovl[base + t] = vbuf[(size_t)row * HDD + t];
}

// ============================================================
// KV sliding-window shift: out[..,0:124,:] = past[..,4:128,:] for K and V,
// all layers at once. One thread per element.
// ============================================================
__global__ void kv_shift(const float* __restrict__ pk, const float* __restrict__ pv,
                         float* __restrict__ ok, float* __restrict__ ov)
{
    const size_t total = (size_t)LL * BB * KVHH * (WW - NN) * HDD;
    size_t idx = (size_t)blockIdx.x * 256 + threadIdx.x;
    if (idx >= total) return;
    int hd = (int)(idx % HDD);
    size_t tmp = idx / HDD;
    int w = (int)(tmp % (WW - NN));
    tmp /= (WW - NN);                          // tmp = ((l*B+b)*KVH+kvh)
    size_t src = (tmp * WW + (w + NN)) * HDD + hd;
    size_t dst = (tmp * WW + w) * HDD + hd;
    ok[dst] = pk[src];
    ov[dst] = pv[src];
}

// ============================================================
// Sliding-window GQA attention: one block per (b, head), 128 threads.
// q,o: [32, NH*HD]; kc,vc: [B,KVH,W,HD] (this layer's cache region).
// Mask: key w valid iff w <= 124 + qi.
// ============================================================
__global__ __launch_bounds__(128) void attn_kernel(
    const float* __restrict__ q, const float* __restrict__ kc,
    const float* __restrict__ vc, float* __restrict__ o)
{
    __shared__ float qs[NN][HDD];
    __shared__ float pr[NN][WW];
    __shared__ float mx[NN];
    __shared__ float sm[NN];
    const int b   = blockIdx.x >> 4;
    const int h   = blockIdx.x & 15;
    const int kvh = h >> 1;                    // rep = NH/KVH = 2
    const int t   = threadIdx.x;

    for (int idx = t; idx < NN * HDD; idx += 128) {
        int qi = idx >> 6, d = idx & 63;
        qs[qi][d] = q[((size_t)(b * NN + qi) * NHH + h) * HDD + d];
    }
    __syncthreads();

    // scores: thread t = key position w
    const float* kp = kc + ((size_t)b * KVHH + kvh) * WW * HDD + (size_t)t * HDD;
    float s[NN] = {0.f, 0.f, 0.f, 0.f};
    for (int d = 0; d < HDD; ++d) {
        float kv = kp[d];
#pragma unroll
        for (int qi = 0; qi < NN; ++qi) s[qi] += kv * qs[qi][d];
    }
    const float scale = 0.125f;                // 1/sqrt(64)
#pragma unroll
    for (int qi = 0; qi < NN; ++qi)
        pr[qi][t] = s[qi] * scale + ((t <= (WW - NN) + qi) ? 0.f : -10000.f);
    __syncthreads();

    if (t < NN) {
        float m = -1e30f;
        for (int w = 0; w < WW; ++w) m = fmaxf(m, pr[t][w]);
        mx[t] = m;
    }
    __syncthreads();
#pragma unroll
    for (int qi = 0; qi < NN; ++qi) pr[qi][t] = __expf(pr[qi][t] - mx[qi]);
    __syncthreads();
    if (t < NN) {
        float ssum = 0.f;
        for (int w = 0; w < WW; ++w) ssum += pr[t][w];
        sm[t] = 1.f / ssum;
    }
    __syncthreads();

    // output: thread handles dim d for two queries (coalesced over v rows)
    const int d  = t & 63;
    const int qb = (t >> 6) * 2;
    const float* vp = vc + ((size_t)b * KVHH + kvh) * WW * HDD;
    float o0 = 0.f, o1 = 0.f;
    for (int w = 0; w < WW; ++w) {
        float vv = vp[(size_t)w * HDD + d];
        o0 += pr[qb][w] * vv;
        o1 += pr[qb + 1][w] * vv;
    }
    o[((size_t)(b * NN + qb) * NHH + h) * HDD + d]     = o0 * sm[qb];
    o[((size_t)(b * NN + qb + 1) * NHH + h) * HDD + d] = o1 * sm[qb + 1];
}

// ============================================================
// SwiGLU: g = silu(g) * u  (in place over g)
// ============================================================
__global__ void swiglu_kernel(float* __restrict__ g, const float* __restrict__ u, int n)
{
    int i = blockIdx.x * 256 + threadIdx.x;
    if (i < n) {
        float x = g[i];
        g[i] = (x / (1.f + __expf(-x))) * u[i];
    }
}

// ============================================================
// Final transposed store: out[b, col, n] = yT[b*4+n, col]
// ============================================================
__global__ void store_newhidden(const float* __restrict__ yT, float* __restrict__ out)
{
    int idx = blockIdx.x * 256 + threadIdx.x;
    if (idx >= MM * DOUTT) return;
    int row = idx / DOUTT, col = idx % DOUTT;
    int b = row >> 2, n = row & 3;
    out[((size_t)b * DOUTT + col) * NN + n] = yT[idx];
}

// ============================================================
// Host orchestration
// ============================================================
extern "C" void kernel_launch(void* const* d_in, const int* in_sizes, int n_in,
                              void* d_out, int out_size, void* d_ws, size_t ws_size,
                              hipStream_t stream)
{
    (void)in_sizes; (void)n_in; (void)out_size; (void)ws_size;
    const float* hidden = (const float*)d_in[0];
    const float* past_k = (const float*)d_in[1];
    const float* past_v = (const float*)d_in[2];
    const float* Win    = (const float*)d_in[3];
    const float* Wout   = (const float*)d_in[4];
    const float* norm_w = (const float*)d_in[5];
    const float* Wq     = (const float*)d_in[6];
    const float* Wk     = (const float*)d_in[7];
    const float* Wv     = (const float*)d_in[8];
    const float* Wo     = (const float*)d_in[9];
    const float* ln1    = (const float*)d_in[10];
    const float* ln2    = (const float*)d_in[11];
    const float* qn     = (const float*)d_in[12];
    const float* kn     = (const float*)d_in[13];
    const float* Wg     = (const float*)d_in[14];
    const float* Wu     = (const float*)d_in[15];
    const float* Wd     = (const float*)d_in[16];

    float* out     = (float*)d_out;
    float* out_hid = out;                                      // 32768
    float* out_ks  = out + (size_t)MM * DOUTT;                 // L*B*KVH*W*HD
    const size_t kvTot  = (size_t)LL * BB * KVHH * WW * HDD;   // 4194304
    float* out_vs  = out_ks + kvTot;
    const size_t cacheL = (size_t)BB * KVHH * WW * HDD;        // 524288 per layer

    // workspace layout (floats), ~1.5 MB total.
    // qb/kb/vb are contiguous (one zero-fill); gb/ub are contiguous too.
    float* ws = (float*)d_ws;
    float* h  = ws;                 // 32768
    float* x  = h  + 32768;         // 32768
    float* qb = x  + 32768;         // 32768
    float* kb = qb + 32768;         // 16384
    float* vb = kb + 16384;         // 16384
    float* ob = vb + 16384;         // 32768
    float* gb = ob + 32768;         // 98304
    float* ub = gb + 98304;         // 98304
    float* yT = ub + 98304;         // 32768

    auto gemm = [&](const float* X, const float* Wm, float* Y, int K, int Nc) {
        dim3 grid(Nc / 64, K / 128);
        gemm32_wmma_f32<<<grid, dim3(128), 0, stream>>>(X, Wm, Y, K, Nc);
    };
    auto zero = [&](float* p, int n) {
        fillf<<<dim3((n + 255) / 256), dim3(256), 0, stream>>>(p, 0.f, n);
    };

    // 1) KV-cache sliding-window shift for all layers (pure copy)
    {
        const size_t total = (size_t)LL * BB * KVHH * (WW - NN) * HDD;
        kv_shift<<<dim3((unsigned)((total + 255) / 256)), dim3(256), 0, stream>>>(
            past_k, past_v, out_ks, out_vs);
    }

    // 2) h = hidden @ Win
    zero(h, MM * DD);
    gemm(hidden, Win, h, HIN, DD);

    // 3) layers
    for (int l = 0; l < LL; ++l) {
        rmsnorm32<<<dim3(MM), dim3(256), 0, stream>>>(h, ln1 + (size_t)l * DD, x, DD);

        zero(qb, 32768 + 16384 + 16384);
        gemm(x, Wq + (size_t)l * DD * (NHH * HDD),  qb, DD, NHH * HDD);
        gemm(x, Wk + (size_t)l * DD * (KVHH * HDD), kb, DD, KVHH * HDD);
        gemm(x, Wv + (size_t)l * DD * (KVHH * HDD), vb, DD, KVHH * HDD);

        rope_q<<<dim3(MM * NHH), dim3(64), 0, stream>>>(qb, qn + (size_t)l * HDD);
        rope_kv<<<dim3(MM * KVHH), dim3(64), 0, stream>>>(
            kb, vb, kn + (size_t)l * HDD, out_ks + (size_t)l * cacheL,
            out_vs + (size_t)l * cacheL);

        attn_kernel<<<dim3(BB * NHH), dim3(128), 0, stream>>>(
            qb, out_ks + (size_t)l * cacheL, out_vs + (size_t)l * cacheL, ob);

        // h += o @ Wo  (residual free via split-K atomic accumulation)
        gemm(ob, Wo + (size_t)l * (NHH * HDD) * DD, h, NHH * HDD, DD);

        rmsnorm32<<<dim3(MM), dim3(256), 0, stream>>>(h, ln2 + (size_t)l * DD, x, DD);

        zero(gb, 2 * MM * FFF);
        gemm(x, Wg + (size_t)l * DD * FFF, gb, DD, FFF);
        gemm(x, Wu + (size_t)l * DD * FFF, ub, DD, FFF);
        swiglu_kernel<<<dim3((MM * FFF + 255) / 256), dim3(256), 0, stream>>>(
            gb, ub, MM * FFF);

        // h += swiglu @ Wd
        gemm(gb, Wd + (size_t)l * FFF * DD, h, FFF, DD);
    }

    // 4) final norm + output projection + transposed store
    rmsnorm32<<<dim3(MM), dim3(256), 0, stream>>>(h, norm_w, x, DD);
    zero(yT, MM * DOUTT);
    gemm(x, Wout, yT, DD, DOUTT);
    store_newhidden<<<dim3((MM * DOUTT + 255) / 256), dim3(256), 0, stream>>>(yT, out_hid);
}